// PatchSelect_49452253446255
// MI455X (gfx1250) — compile-verified
//
#include <hip/hip_runtime.h>
#include <hip/hip_bf16.h>

// ---------------------------------------------------------------------------
// PatchSelect on MI455X (gfx1250, wave32).
//
// Per patch (one 256-thread / 8-wave workgroup each, 51076 blocks):
//   1) hybrid bitonic sort of the 48x48 window (padded to 4096 with +inf):
//      - 16 elements/thread sorted in REGISTERS (static min/max network)
//      - only j>=16 merge steps touch LDS; j=8..1 finish in registers
//      => ~40% less LDS traffic and ~half the barriers vs pure-LDS bitonic
//   2) 1-Wasserstein distance vs the pre-sorted query via chained
//      V_WMMA_F32_16X16X4_F32 (A = |diff|, B = ones -> exact f32 row sums)
//   3) per-batch argmin via packed (float_bits<<32 | index) u64 atomicMin
// ---------------------------------------------------------------------------

typedef __attribute__((ext_vector_type(2))) float v2f;
typedef __attribute__((ext_vector_type(8))) float v8f;

#define QW 48
#define KH 160
#define KW 160
#define NPIX 2304                 // 48*48 pixels per patch
#define NSORT 4096                // next pow2, padded with +inf
#define P_OUT 113                 // 160 - 48 + 1
#define L_PATCH (P_OUT * P_OUT)   // 12769 patches per batch
#define NBATCH 4
#define NTHREADS 256              // 8 waves of 32
#define ELEMS 16                  // register block per thread (16*256 = 4096)
#define NBLK (NPIX / ELEMS)       // 144 threads carry real data
#define NCHUNK (NPIX / 64)        // 36 WMMA chunks of 64 diffs

__device__ __forceinline__ void cexch(float& a, float& b, bool up) {
    float lo = fminf(a, b);
    float hi = fmaxf(a, b);
    a = up ? lo : hi;
    b = up ? hi : lo;
}

// Static bitonic sorting network: 16 elements ascending, fully unrolled,
// all directions compile-time -> pure v_min/v_max VALU, no LDS.
__device__ __forceinline__ void sort16_asc(float (&r)[ELEMS]) {
#pragma unroll
    for (int k = 2; k <= ELEMS; k <<= 1) {
#pragma unroll
        for (int j = k >> 1; j > 0; j >>= 1) {
#pragma unroll
            for (int l = 0; l < ELEMS; ++l) {
                int p = l ^ j;
                if (p > l) cexch(r[l], r[p], (l & k) == 0);  // l&16==0 always
            }
        }
    }
}

// Bitonic merge of a 16-element block, direction 'up' (j = 8,4,2,1).
__device__ __forceinline__ void merge16(float (&r)[ELEMS], bool up) {
#pragma unroll
    for (int j = ELEMS / 2; j > 0; j >>= 1) {
#pragma unroll
        for (int l = 0; l < ELEMS; ++l) {
            int p = l ^ j;
            if (p > l) cexch(r[l], r[p], up);
        }
    }
}

// Hybrid bitonic sort of 4096 floats: r[] holds thread tid's elements
// [16*tid, 16*tid+16) on entry (unsorted); s[] ascending-sorted on exit.
// Must be called by all 256 threads (contains barriers).
__device__ __forceinline__ void hybrid_sort_4096(float* s, int tid,
                                                 float (&r)[ELEMS]) {
    // Phase A (k = 2..16) entirely in registers: ascending sort, then
    // reverse odd blocks so block t is sorted ascending iff (i&16)==0.
    sort16_asc(r);
    if (tid & 1) {
#pragma unroll
        for (int l = 0; l < ELEMS / 2; ++l) {
            float t = r[l]; r[l] = r[ELEMS - 1 - l]; r[ELEMS - 1 - l] = t;
        }
    }
#pragma unroll
    for (int l = 0; l < ELEMS; ++l) s[tid * ELEMS + l] = r[l];
    __syncthreads();

    // Phases k = 32..4096: j >= 16 cross-block steps in LDS,
    // j = 8..1 finish as a register merge (direction uniform per block).
    for (int k = 2 * ELEMS; k <= NSORT; k <<= 1) {
        for (int j = k >> 1; j >= ELEMS; j >>= 1) {
#pragma unroll
            for (int u = 0; u < NSORT / 2 / NTHREADS; ++u) {
                int t2 = tid + u * NTHREADS;
                int i = ((t2 & ~(j - 1)) << 1) | (t2 & (j - 1));
                int p = i + j;
                bool up = ((i & k) == 0);
                float a = s[i];
                float b = s[p];
                if ((a > b) == up) { s[i] = b; s[p] = a; }
            }
            __syncthreads();
        }
        bool up = (((tid * ELEMS) & k) == 0);   // i&k constant within block
#pragma unroll
        for (int l = 0; l < ELEMS; ++l) r[l] = s[tid * ELEMS + l];
        merge16(r, up);
#pragma unroll
        for (int l = 0; l < ELEMS; ++l) s[tid * ELEMS + l] = r[l];
        __syncthreads();
    }
}

// Kernel 1: sort each batch's query patch into d_ws and reset the argmin
// slots (must happen every launch: harness does not restore ws between
// graph replays).
__global__ __launch_bounds__(NTHREADS)
void ps_sort_query_kernel(const float* __restrict__ query,
                          float* __restrict__ sortedQ,
                          unsigned long long* __restrict__ best) {
    __shared__ float s[NSORT];
    const int b = blockIdx.x;
    const int tid = threadIdx.x;

    float r[ELEMS];
    if (tid < NBLK) {
        const float* src = query + b * NPIX + tid * ELEMS;  // contiguous
#pragma unroll
        for (int l = 0; l < ELEMS; ++l) r[l] = src[l];
    } else {
#pragma unroll
        for (int l = 0; l < ELEMS; ++l) r[l] = __builtin_inff();
    }
    if (tid == 0) best[b] = 0xFFFFFFFFFFFFFFFFull;

    hybrid_sort_4096(s, tid, r);

    for (int i = tid; i < NPIX; i += NTHREADS)
        sortedQ[b * NPIX + i] = s[i];
}

// Kernel 2: one workgroup per (batch, patch).
__global__ __launch_bounds__(NTHREADS)
void ps_match_kernel(const float* __restrict__ key,
                     const float* __restrict__ sortedQ,
                     unsigned long long* __restrict__ best) {
    __shared__ float s[NSORT];
    __shared__ float q[NPIX];
    __shared__ float wsum[NTHREADS / 32];

    const unsigned gbid = blockIdx.x;
    const int b    = (int)(gbid / L_PATCH);
    const int pIdx = (int)(gbid % L_PATCH);
    const int ho   = pIdx / P_OUT;
    const int wo   = pIdx % P_OUT;
    const int tid  = threadIdx.x;

    // Each 16-block lies within one image row (48 = 3*16): contiguous loads.
    float r[ELEMS];
    const float* kb = key + (size_t)b * KH * KW;
    if (tid < NBLK) {
        int row  = tid / 3;
        int col0 = (tid - row * 3) * ELEMS;
        const float* src = kb + (ho + row) * KW + (wo + col0);
#pragma unroll
        for (int l = 0; l < ELEMS; ++l) r[l] = src[l];
    } else {
#pragma unroll
        for (int l = 0; l < ELEMS; ++l) r[l] = __builtin_inff();
    }
    // Sorted query for this batch -> LDS (L2-hot; reused by 12769 blocks).
    for (int i = tid; i < NPIX; i += NTHREADS)
        q[i] = sortedQ[b * NPIX + i];

    hybrid_sort_4096(s, tid, r);
    // (sort ends with a barrier; s[0..2303] ascending, q[] loaded)

    // --- WMMA L1 reduction ------------------------------------------------
    // 36 chunks of 64 diffs round-robin over 8 waves (wave-uniform: EXEC is
    // all-ones at every WMMA). A (16x4 f32, 2 VGPRs/lane) holds |diff|,
    // B = ones => D[m][n] = sum_k A[m][k], chained through C in exact f32.
    const int w    = tid >> 5;
    const int lane = tid & 31;

    v8f acc = {};
    v2f ones;
    ones.x = 1.0f;
    ones.y = 1.0f;

    for (int c = w; c < NCHUNK; c += NTHREADS / 32) {
        int idx = c * 64 + lane * 2;     // bijection of 64 slots -> (m,k)
        v2f a;
        a.x = fabsf(s[idx]     - q[idx]);
        a.y = fabsf(s[idx + 1] - q[idx + 1]);
        // (neg_a, A, neg_b, B, c_mod, C, reuse_a, reuse_b)
        acc = __builtin_amdgcn_wmma_f32_16x16x4_f32(
            false, a, false, ones, (short)0, acc, false, false);
    }

    // Lane-local sum of 8 D VGPRs = half-matrix sum; shfl_xor(16) combines
    // the M=0..7 / M=8..15 halves (wave32).
    float t = acc[0] + acc[1] + acc[2] + acc[3] +
              acc[4] + acc[5] + acc[6] + acc[7];
    t += __shfl_xor(t, 16, 32);
    if (lane == 0) wsum[w] = t;
    __syncthreads();

    if (tid == 0) {
        float tot = 0.0f;
#pragma unroll
        for (int i = 0; i < NTHREADS / 32; ++i) tot += wsum[i];
        float y = tot * (1.0f / (float)NPIX);   // mean over pixels (C==1)
        // y >= 0 => IEEE bits order-preserving as unsigned; low-word index
        // makes ties resolve to the lowest index (matches jnp.argmin).
        unsigned long long keyv =
            ((unsigned long long)__float_as_uint(y) << 32) | (unsigned)pIdx;
        atomicMin(&best[b], keyv);
    }
}

// Kernel 3: decode argmin indices, emit [idx0..idx3, P] as floats.
__global__ void ps_finalize_kernel(const unsigned long long* __restrict__ best,
                                   float* __restrict__ out, int out_size) {
    int i = threadIdx.x;
    if (i < NBATCH && i < out_size)
        out[i] = (float)(unsigned)(best[i] & 0xFFFFFFFFull);
    if (i == NBATCH && out_size > NBATCH)
        out[NBATCH] = (float)P_OUT;
}

extern "C" void kernel_launch(void* const* d_in, const int* in_sizes, int n_in,
                              void* d_out, int out_size, void* d_ws, size_t ws_size,
                              hipStream_t stream) {
    const float* query = (const float*)d_in[0];   // (4,1,48,48) f32
    const float* key   = (const float*)d_in[1];   // (4,1,160,160) f32

    // Workspace layout: [0,32): 4x u64 argmin slots; [64,...): sorted queries.
    unsigned long long* best = (unsigned long long*)d_ws;
    float* sortedQ = (float*)((char*)d_ws + 64);

    ps_sort_query_kernel<<<NBATCH, NTHREADS, 0, stream>>>(query, sortedQ, best);
    ps_match_kernel<<<NBATCH * L_PATCH, NTHREADS, 0, stream>>>(key, sortedQ, best);
    ps_finalize_kernel<<<1, 8, 0, stream>>>(best, (float*)d_out, out_size);
}